// Encoder_27169963114977
// MI455X (gfx1250) — compile-verified
//
#include <hip/hip_runtime.h>

#define N_NODES 100000
#define N_EDGES 1600000
#define DIM 128
#define LAYERS 3

typedef __attribute__((ext_vector_type(2))) float v2f;
typedef __attribute__((ext_vector_type(8))) float v8f;

__global__ void zero_f32(float* __restrict__ p, int n) {
  int i = blockIdx.x * blockDim.x + threadIdx.x;
  int stride = gridDim.x * blockDim.x;
  for (; i < n; i += stride) p[i] = 0.0f;
}

// deg[dst] += 1 for every edge (float so the invert pass is trivial)
__global__ void deg_count(const long long* __restrict__ ei, float* __restrict__ deg) {
  int e = blockIdx.x * blockDim.x + threadIdx.x;
  if (e < N_EDGES) {
    int d = (int)ei[N_EDGES + e];  // dst row of edge_index
    atomicAdd(&deg[d], 1.0f);
  }
}

__global__ void deg_invert(float* __restrict__ deg) {
  int i = blockIdx.x * blockDim.x + threadIdx.x;
  if (i < N_NODES) {
    float d = deg[i];
    deg[i] = d > 0.0f ? 1.0f / d : 0.0f;  // == where(deg>0, 1/max(deg,1), 0)
  }
}

// One wave (32 lanes) per edge; each lane moves 4 contiguous floats.
// Gather z[src] (coalesced 512B) and atomically add into aggr[dst].
__global__ void scatter_add(const float* __restrict__ z,
                            const long long* __restrict__ ei,
                            float* __restrict__ aggr) {
  int t = blockIdx.x * blockDim.x + threadIdx.x;
  int e = t >> 5;
  int f = (t & 31) << 2;
  if (e >= N_EDGES) return;
  int s = (int)ei[e];
  int d = (int)ei[N_EDGES + e];
  const float4 v = *(const float4*)(z + (size_t)s * DIM + f);
  float* dp = aggr + (size_t)d * DIM + f;
  atomicAdd(dp + 0, v.x);
  atomicAdd(dp + 1, v.y);
  atomicAdd(dp + 2, v.z);
  atomicAdd(dp + 3, v.w);
}

// Fused: out = relu( (aggr*deg_inv) @ Wl + z @ Wr + bl )
// Block = 8 waves = one 16-row strip; wave w handles column tile [16w, 16w+16).
// V_WMMA_F32_16X16X4_F32, K stepped by 4, two WMMAs per step (Wl and Wr paths).
__global__ __launch_bounds__(256) void sage_gemm(
    const float* __restrict__ aggr, const float* __restrict__ deg_inv,
    const float* __restrict__ z,
    const float* __restrict__ Wl, const float* __restrict__ bl,
    const float* __restrict__ Wr, float* __restrict__ out) {
  const int wave = threadIdx.x >> 5;
  const int lane = threadIdx.x & 31;
  const int l16  = lane & 15;
  const int kh   = lane >> 4;            // 0: K0/K1 half, 1: K2/K3 half (A frag)
  const int row0 = blockIdx.x * 16;
  const int col0 = wave * 16;

  const int arow = row0 + l16;           // A-matrix: lane <-> row M
  const float dinv = deg_inv[arow];
  const float* pa = aggr + (size_t)arow * DIM;
  const float* pz = z    + (size_t)arow * DIM;

  v8f acc = {};
#pragma unroll 4
  for (int k0 = 0; k0 < DIM; k0 += 4) {
    const int ka = k0 + 2 * kh;
    // A frag (16x4 f32): v0 = K{0|2}, v1 = K{1|3} per ISA layout; fold deg_inv
    v2f aA, aZ, bL, bR;
    aA.x = pa[ka] * dinv;  aA.y = pa[ka + 1] * dinv;
    aZ.x = pz[ka];         aZ.y = pz[ka + 1];
    // B frag (4x16 f32): row K striped across lanes (symmetric K split)
    bL.x = Wl[(size_t)ka * DIM + col0 + l16];
    bL.y = Wl[(size_t)(ka + 1) * DIM + col0 + l16];
    bR.x = Wr[(size_t)ka * DIM + col0 + l16];
    bR.y = Wr[(size_t)(ka + 1) * DIM + col0 + l16];
    acc = __builtin_amdgcn_wmma_f32_16x16x4_f32(false, aA, false, bL,
                                                (short)0, acc, false, false);
    acc = __builtin_amdgcn_wmma_f32_16x16x4_f32(false, aZ, false, bR,
                                                (short)0, acc, false, false);
  }

  const float bias = bl[col0 + l16];
#pragma unroll
  for (int r = 0; r < 8; ++r) {
    // C/D layout: VGPR r = row r (lanes 0-15) / row r+8 (lanes 16-31), N = lane
    float v = acc[r] + bias;
    v = v > 0.0f ? v : 0.0f;
    out[(size_t)(row0 + r + 8 * kh) * DIM + col0 + l16] = v;
  }
}

extern "C" void kernel_launch(void* const* d_in, const int* in_sizes, int n_in,
                              void* d_out, int out_size, void* d_ws, size_t ws_size,
                              hipStream_t stream) {
  const float*     x  = (const float*)d_in[0];
  const long long* ei = (const long long*)d_in[1];  // int64 edge_index [2, E]
  const float*     Wl = (const float*)d_in[2];
  const float*     bl = (const float*)d_in[3];
  const float*     Wr = (const float*)d_in[4];
  float*           out = (float*)d_out;

  // Workspace layout: deg_inv[N] | aggr[N*D] | zA[N*D]  (~103 MB)
  float* deg_inv = (float*)d_ws;
  float* aggr    = deg_inv + N_NODES;
  float* zA      = aggr + (size_t)N_NODES * DIM;

  // Degree (computed once)
  zero_f32<<<1024, 256, 0, stream>>>(deg_inv, N_NODES);
  deg_count<<<(N_EDGES + 255) / 256, 256, 0, stream>>>(ei, deg_inv);
  deg_invert<<<(N_NODES + 255) / 256, 256, 0, stream>>>(deg_inv);

  // Layer ping-pong: x -> out -> zA -> out (no aliasing within a layer)
  for (int layer = 0; layer < LAYERS; ++layer) {
    const float* zin  = (layer == 0) ? x : (layer == 1 ? out : zA);
    float*       zout = (layer == 1) ? zA : out;

    zero_f32<<<4096, 256, 0, stream>>>(aggr, N_NODES * DIM);
    scatter_add<<<N_EDGES / 8, 256, 0, stream>>>(zin, ei, aggr);
    sage_gemm<<<N_NODES / 16, 256, 0, stream>>>(
        aggr, deg_inv, zin,
        Wl + (size_t)layer * DIM * DIM, bl + (size_t)layer * DIM,
        Wr + (size_t)layer * DIM * DIM, zout);
  }
}